// NeighborhoodAttention_2302102471280
// MI455X (gfx1250) — compile-verified
//
#include <hip/hip_runtime.h>

typedef __attribute__((ext_vector_type(2))) float v2f;
typedef __attribute__((ext_vector_type(8))) float v8f;

#define HW    4096
#define CDIM  256
#define HDIM  32
#define QSCALE 0.17677669529663687f   // 32^-0.5

__device__ __forceinline__ v8f wmma_f32_k4(v2f a, v2f b, v8f c) {
  // D = A(16x4 f32) * B(4x16 f32) + C(16x16 f32)
  return __builtin_amdgcn_wmma_f32_16x16x4_f32(
      /*neg_a=*/false, a, /*neg_b=*/false, b,
      /*c_mod=*/(short)0, c, /*reuse_a=*/false, /*reuse_b=*/false);
}

// ---------------------------------------------------------------------------
// Kernel 1: fused QKV GEMM.
// Computes only the head-sliced rows the reference actually uses:
//   qkv[(b*768 + t*256 + hg*32 + d) * HW + p]
//     = sum_c W_{range(hg)}[t*256 + hg*32 + d, c] * x[b, c, p]  (+ bias, *scale for q)
// One wave -> 16 rows x 64 pixels, K-loop of 64 WMMA f32 16x16x4.
// blockDim = 128 (4 waves), grid = (48/4, 4096/64, B) = (12, 64, 2)
// ---------------------------------------------------------------------------
__global__ void qkv_gemm(const float* __restrict__ x,
                         const float* __restrict__ w0, const float* __restrict__ b0,
                         const float* __restrict__ w1, const float* __restrict__ b1,
                         const float* __restrict__ w2, const float* __restrict__ b2,
                         float* __restrict__ qkv) {
  const int wave  = threadIdx.x >> 5;
  const int lane  = threadIdx.x & 31;
  const int mtile = blockIdx.x * 4 + wave;   // 0..47
  const int m0    = mtile * 16;              // row base in [0,768)
  const int pbase = blockIdx.y * 64;
  const int b     = blockIdx.z;

  // head & range selection (heads 0-3 -> range0, 4-6 -> range1, 7 -> range2)
  const int hg = (m0 & 255) >> 5;
  const float* W; const float* bias;
  if (hg < 4)      { W = w0; bias = b0; }
  else if (hg < 7) { W = w1; bias = b1; }
  else             { W = w2; bias = b2; }
  const int t = m0 >> 8;   // 0=q, 1=k, 2=v

  const int lrow = lane & 15;
  const int koff = (lane >> 4) << 1;         // lanes 0-15: K+0/K+1, lanes 16-31: K+2/K+3
  const float* wrow = W + (size_t)(m0 + lrow) * CDIM;
  const float* xb   = x + (size_t)b * CDIM * HW;

  v8f c0 = {}, c1 = {}, c2 = {}, c3 = {};
  for (int k = 0; k < CDIM; k += 4) {
    v2f a;
    a.x = wrow[k + koff];
    a.y = wrow[k + koff + 1];
    const float* xk = xb + (size_t)(k + koff) * HW + pbase + lrow;
    v2f bb0, bb1, bb2, bb3;
    bb0.x = xk[0];   bb0.y = xk[HW];
    bb1.x = xk[16];  bb1.y = xk[HW + 16];
    bb2.x = xk[32];  bb2.y = xk[HW + 32];
    bb3.x = xk[48];  bb3.y = xk[HW + 48];
    c0 = wmma_f32_k4(a, bb0, c0);
    c1 = wmma_f32_k4(a, bb1, c1);
    c2 = wmma_f32_k4(a, bb2, c2);
    c3 = wmma_f32_k4(a, bb3, c3);
  }

  // C/D layout: VGPR r, lanes 0-15 -> M=r, N=lane; lanes 16-31 -> M=r+8, N=lane-16
  const int mhi = (lane >> 4) << 3;
  const float qs = (t == 0) ? QSCALE : 1.0f;
  float* outB = qkv + (size_t)(b * 768) * HW;
#pragma unroll
  for (int r = 0; r < 8; ++r) {
    const int m  = r + mhi;
    const float bv = bias[m0 + m];
    const size_t row = (size_t)(m0 + m) * HW;
    outB[row + pbase      + lrow] = (c0[r] + bv) * qs;
    outB[row + pbase + 16 + lrow] = (c1[r] + bv) * qs;
    outB[row + pbase + 32 + lrow] = (c2[r] + bv) * qs;
    outB[row + pbase + 48 + lrow] = (c3[r] + bv) * qs;
  }
}

// ---------------------------------------------------------------------------
// Kernel 2: neighborhood attention, online softmax, one thread per (b,head,p).
// Reference zero-pads: OOB neighbors contribute logit 0 (and v = 0).
// grid = (4096/256, 8, 2), blockDim = 256
// ---------------------------------------------------------------------------
__global__ void natten(const float* __restrict__ qkv, float* __restrict__ attn_out) {
  const int p  = blockIdx.x * blockDim.x + threadIdx.x;  // 0..4095
  const int hg = blockIdx.y;
  const int b  = blockIdx.z;
  const int py = p >> 6, px = p & 63;

  int ksz, dil;
  if (hg < 4)      { ksz = 5; dil = 1; }
  else if (hg < 7) { ksz = 7; dil = 2; }
  else             { ksz = 9; dil = 3; }
  const int rad = ksz >> 1;

  const size_t qbase = ((size_t)(b * 768)       + hg * HDIM) * HW;
  const size_t kbase = ((size_t)(b * 768) + 256 + hg * HDIM) * HW;
  const size_t vbase = ((size_t)(b * 768) + 512 + hg * HDIM) * HW;

  float qreg[HDIM];
#pragma unroll
  for (int d = 0; d < HDIM; ++d) qreg[d] = qkv[qbase + (size_t)d * HW + p];

  float mx = -1e30f, ssum = 0.0f;
  float acc[HDIM];
#pragma unroll
  for (int d = 0; d < HDIM; ++d) acc[d] = 0.0f;

  for (int ky = -rad; ky <= rad; ++ky) {
    const int ny = py + ky * dil;
    for (int kx = -rad; kx <= rad; ++kx) {
      const int nx = px + kx * dil;
      const bool inb = ((unsigned)ny < 64u) && ((unsigned)nx < 64u);
      const int np = (ny << 6) + nx;
      float s = 0.0f;
      if (inb) {
#pragma unroll
        for (int d = 0; d < HDIM; ++d)
          s = fmaf(qreg[d], qkv[kbase + (size_t)d * HW + np], s);
      }
      const float mn   = fmaxf(mx, s);
      const float corr = __expf(mx - mn);
      const float w    = __expf(s - mn);
      ssum = ssum * corr + w;
      if (inb) {
#pragma unroll
        for (int d = 0; d < HDIM; ++d)
          acc[d] = fmaf(acc[d], corr, w * qkv[vbase + (size_t)d * HW + np]);
      } else {
#pragma unroll
        for (int d = 0; d < HDIM; ++d) acc[d] *= corr;
      }
      mx = mn;
    }
  }

  const float inv = 1.0f / ssum;
  const size_t obase = ((size_t)(b * 256) + hg * HDIM) * HW;
#pragma unroll
  for (int d = 0; d < HDIM; ++d)
    attn_out[obase + (size_t)d * HW + p] = acc[d] * inv;
}

// ---------------------------------------------------------------------------
// Kernel 3: output projection GEMM (256x256 x 256xHW per batch) via WMMA f32.
// grid = (16/4, 64, 2), blockDim = 128
// ---------------------------------------------------------------------------
__global__ void proj_gemm(const float* __restrict__ attn,
                          const float* __restrict__ pw,
                          const float* __restrict__ pb,
                          float* __restrict__ out) {
  const int wave  = threadIdx.x >> 5;
  const int lane  = threadIdx.x & 31;
  const int m0    = (blockIdx.x * 4 + wave) * 16;  // [0,256)
  const int pbase = blockIdx.y * 64;
  const int b     = blockIdx.z;

  const int lrow = lane & 15;
  const int koff = (lane >> 4) << 1;
  const float* wrow = pw + (size_t)(m0 + lrow) * CDIM;
  const float* ab   = attn + (size_t)(b * 256) * HW;

  v8f c0 = {}, c1 = {}, c2 = {}, c3 = {};
  for (int k = 0; k < CDIM; k += 4) {
    v2f a;
    a.x = wrow[k + koff];
    a.y = wrow[k + koff + 1];
    const float* xk = ab + (size_t)(k + koff) * HW + pbase + lrow;
    v2f bb0, bb1, bb2, bb3;
    bb0.x = xk[0];   bb0.y = xk[HW];
    bb1.x = xk[16];  bb1.y = xk[HW + 16];
    bb2.x = xk[32];  bb2.y = xk[HW + 32];
    bb3.x = xk[48];  bb3.y = xk[HW + 48];
    c0 = wmma_f32_k4(a, bb0, c0);
    c1 = wmma_f32_k4(a, bb1, c1);
    c2 = wmma_f32_k4(a, bb2, c2);
    c3 = wmma_f32_k4(a, bb3, c3);
  }

  const int mhi = (lane >> 4) << 3;
  float* outB = out + (size_t)(b * 256) * HW;
#pragma unroll
  for (int r = 0; r < 8; ++r) {
    const int m  = r + mhi;
    const float bv = pb[m0 + m];
    const size_t row = (size_t)(m0 + m) * HW;
    outB[row + pbase      + lrow] = c0[r] + bv;
    outB[row + pbase + 16 + lrow] = c1[r] + bv;
    outB[row + pbase + 32 + lrow] = c2[r] + bv;
    outB[row + pbase + 48 + lrow] = c3[r] + bv;
  }
}

extern "C" void kernel_launch(void* const* d_in, const int* in_sizes, int n_in,
                              void* d_out, int out_size, void* d_ws, size_t ws_size,
                              hipStream_t stream) {
  (void)in_sizes; (void)n_in; (void)out_size; (void)ws_size;
  const float* x  = (const float*)d_in[0];
  const float* w0 = (const float*)d_in[1];
  const float* b0 = (const float*)d_in[2];
  const float* w1 = (const float*)d_in[3];
  const float* b1 = (const float*)d_in[4];
  const float* w2 = (const float*)d_in[5];
  const float* b2 = (const float*)d_in[6];
  const float* pw = (const float*)d_in[7];
  const float* pb = (const float*)d_in[8];

  float* qkv  = (float*)d_ws;                       // 2*768*4096 f32 = 25.2 MB
  float* attn = qkv + (size_t)2 * 768 * HW;         // 2*256*4096 f32 =  8.4 MB

  qkv_gemm<<<dim3(12, 64, 2), 128, 0, stream>>>(x, w0, b0, w1, b1, w2, b2, qkv);
  natten  <<<dim3(16,  8, 2), 256, 0, stream>>>(qkv, attn);
  proj_gemm<<<dim3(4, 64, 2), 128, 0, stream>>>(attn, pw, pb, (float*)d_out);
}